// GraphSearchPolicy_489626272373
// MI455X (gfx1250) — compile-verified
//
#include <hip/hip_runtime.h>
#include <hip/hip_bf16.h>

// Problem constants (match reference)
#define B_   2048
#define A_   256
#define ED_  200
#define RD_  200
#define HD_  400
#define AD_  400   // ED+RD
#define XD_  800   // ED+HD+RD
#define HUGE_ 1e9f

typedef __attribute__((ext_vector_type(2))) float v2f;
typedef __attribute__((ext_vector_type(8))) float v8f;

// ---------------------------------------------------------------------------
// Kernel 1: build X = [entity_emb[e] | H | relation_emb[q]]  -> [B, 800]
// ---------------------------------------------------------------------------
__global__ __launch_bounds__(256) void build_x_kernel(
    const int* __restrict__ e, const int* __restrict__ q,
    const float* __restrict__ H,
    const float* __restrict__ ent_emb, const float* __restrict__ rel_emb,
    float* __restrict__ X)
{
    const int b = blockIdx.x;
    const float* erow = ent_emb + (size_t)e[b] * ED_;
    const float* qrow = rel_emb + (size_t)q[b] * RD_;
    const float* hrow = H + (size_t)b * HD_;
    float* xrow = X + (size_t)b * XD_;
    for (int i = threadIdx.x; i < ED_; i += 256) xrow[i]             = erow[i];
    for (int i = threadIdx.x; i < HD_; i += 256) xrow[ED_ + i]       = hrow[i];
    for (int i = threadIdx.x; i < RD_; i += 256) xrow[ED_ + HD_ + i] = qrow[i];
}

// ---------------------------------------------------------------------------
// Kernel 2: fp32 GEMM via V_WMMA_F32_16X16X4_F32, fused bias (+ optional ReLU)
//   C[M,N] = act(A[M,K] @ B[K,N] + bias[N]),  N compile-time (offset folding).
//   block = 256 threads = 8 waves; wave w -> rows [bx*128 + w*16, +16).
//   Each wave computes NT adjacent 16-wide N-tiles (16 x 16*NT strip),
//   reusing one A-fragment across NT WMMAs per K-step.
//   Requires M%128==0, N%(16*NT)==0, K%4==0.
// ---------------------------------------------------------------------------
template <int K, int N, int NT, bool RELU>
__global__ __launch_bounds__(256) void gemm_wmma_f32_kernel(
    const float* __restrict__ Amat,   // [M, K] row-major
    const float* __restrict__ Bmat,   // [K, N] row-major
    const float* __restrict__ bias,   // [N]
    float* __restrict__ Cmat)         // [M, N] row-major
{
    const int lane = threadIdx.x & 31;
    const int wave = threadIdx.x >> 5;
    const int m0 = blockIdx.x * 128 + wave * 16;
    const int n0 = blockIdx.y * (16 * NT);

    // fp32 WMMA fragment addressing (ISA 7.12.2):
    //  A (16x4): lanes 0-15 hold row M=lane,    VGPR0=K,   VGPR1=K+1
    //            lanes 16-31 hold row M=lane-16, VGPR0=K+2, VGPR1=K+3
    //  B (4x16): lanes 0-15 col N=lane with K/K+1; lanes 16-31 with K+2/K+3
    const int koff = (lane >> 4) * 2;         // 0 or 2
    const int lcol = lane & 15;

    const float* aptr = Amat + (size_t)(m0 + lcol) * K + koff;   // + k (imm)
    const float* bptr = Bmat + (size_t)koff * N + n0 + lcol;     // + k*N + t*16 (imm)

    v8f acc[NT];
#pragma unroll
    for (int t = 0; t < NT; ++t) acc[t] = (v8f){};

    for (int k = 0; k < K; k += 4) {
        const float2 av = *(const float2*)(aptr + k);
        v2f a;
        a[0] = av.x; a[1] = av.y;
#pragma unroll
        for (int t = 0; t < NT; ++t) {
            v2f b;
            b[0] = bptr[(size_t)k * N + t * 16];
            b[1] = bptr[(size_t)(k + 1) * N + t * 16];
            acc[t] = __builtin_amdgcn_wmma_f32_16x16x4_f32(
                /*neg_a=*/false, a, /*neg_b=*/false, b,
                /*c_mod=*/(short)0, acc[t], /*reuse_a=*/false, /*reuse_b=*/false);
        }
    }

    // D layout: VGPR v -> row m0+v (lanes 0-15) / m0+v+8 (lanes 16-31)
    const int rbase = m0 + (lane >> 4) * 8;
#pragma unroll
    for (int t = 0; t < NT; ++t) {
        const int col = n0 + t * 16 + lcol;
        const float bv = bias[col];
#pragma unroll
        for (int v = 0; v < 8; ++v) {
            float val = acc[t][v] + bv;
            if (RELU) val = fmaxf(val, 0.0f);
            Cmat[(size_t)(rbase + v) * N + col] = val;
        }
    }
}

// ---------------------------------------------------------------------------
// Kernel 3: fused gather-GEMV + mask + softmax + entropy.
//   One block per batch row b; one thread per action a (256 threads).
//   scores[b,a] = dot(rel_emb[r_space[b,a]], X2[b][0:200])
//               + dot(ent_emb[e_space[b,a]], X2[b][200:400]) - (1-mask)*HUGE
// ---------------------------------------------------------------------------
__global__ __launch_bounds__(256) void scores_softmax_kernel(
    const int* __restrict__ r_space, const int* __restrict__ e_space,
    const int* __restrict__ action_mask,
    const float* __restrict__ X2,
    const float* __restrict__ ent_emb, const float* __restrict__ rel_emb,
    float* __restrict__ out_dist,     // [B, A]
    float* __restrict__ out_entropy)  // [B]
{
    __shared__ __align__(16) float x2s[AD_];
    __shared__ float red[256];

    const int b = blockIdx.x;
    const int a = threadIdx.x;

    for (int i = threadIdx.x; i < AD_; i += 256)
        x2s[i] = X2[(size_t)b * AD_ + i];
    __syncthreads();

    const size_t base = (size_t)b * A_ + a;
    const float4* rrow = (const float4*)(rel_emb + (size_t)r_space[base] * RD_);
    const float4* erow = (const float4*)(ent_emb + (size_t)e_space[base] * ED_);
    const float4* xr = (const float4*)(x2s);        // X2[0:200]
    const float4* xe = (const float4*)(x2s + RD_);  // X2[200:400] (800B, 16B aligned)

    float s = 0.0f;
#pragma unroll 5
    for (int i = 0; i < RD_ / 4; ++i) {
        const float4 r4 = rrow[i], x4 = xr[i];
        s += r4.x * x4.x + r4.y * x4.y + r4.z * x4.z + r4.w * x4.w;
    }
#pragma unroll 5
    for (int i = 0; i < ED_ / 4; ++i) {
        const float4 e4 = erow[i], x4 = xe[i];
        s += e4.x * x4.x + e4.y * x4.y + e4.z * x4.z + e4.w * x4.w;
    }
    const float sc = s - (1.0f - (float)action_mask[base]) * HUGE_;

    // block max
    red[a] = sc; __syncthreads();
    for (int off = 128; off > 0; off >>= 1) {
        if (a < off) red[a] = fmaxf(red[a], red[a + off]);
        __syncthreads();
    }
    const float mx = red[0]; __syncthreads();

    // exp & block sum
    const float ex = __expf(sc - mx);
    red[a] = ex; __syncthreads();
    for (int off = 128; off > 0; off >>= 1) {
        if (a < off) red[a] += red[a + off];
        __syncthreads();
    }
    const float inv_sum = 1.0f / red[0]; __syncthreads();

    const float p = ex * inv_sum;
    out_dist[base] = p;

    // entropy = -sum p*log(clip(p,1e-20,1))
    red[a] = -p * __logf(fmaxf(p, 1e-20f)); __syncthreads();
    for (int off = 128; off > 0; off >>= 1) {
        if (a < off) red[a] += red[a + off];
        __syncthreads();
    }
    if (a == 0) out_entropy[b] = red[0];
}

// ---------------------------------------------------------------------------
extern "C" void kernel_launch(void* const* d_in, const int* in_sizes, int n_in,
                              void* d_out, int out_size, void* d_ws, size_t ws_size,
                              hipStream_t stream)
{
    (void)in_sizes; (void)n_in; (void)out_size; (void)ws_size;

    const int*   e           = (const int*)d_in[0];
    const int*   q           = (const int*)d_in[1];
    const int*   r_space     = (const int*)d_in[2];
    const int*   e_space     = (const int*)d_in[3];
    const int*   action_mask = (const int*)d_in[4];
    const float* H           = (const float*)d_in[5];
    const float* ent_emb     = (const float*)d_in[6];
    const float* rel_emb     = (const float*)d_in[7];
    const float* W1          = (const float*)d_in[8];
    const float* b1          = (const float*)d_in[9];
    const float* W2          = (const float*)d_in[10];
    const float* b2          = (const float*)d_in[11];

    float* out_dist    = (float*)d_out;            // [B, A]
    float* out_entropy = out_dist + (size_t)B_ * A_;

    // workspace layout
    float* X  = (float*)d_ws;                      // [B, 800]
    float* h  = X  + (size_t)B_ * XD_;             // [B, 400]
    float* X2 = h  + (size_t)B_ * AD_;             // [B, 400]

    // 1) X = [E | H | Q]
    build_x_kernel<<<B_, 256, 0, stream>>>(e, q, H, ent_emb, rel_emb, X);

    // 2) h = relu(X @ W1 + b1)   (M=2048, K=800, N=400); block tile 128x80
    gemm_wmma_f32_kernel<XD_, AD_, 5, true>
        <<<dim3(B_ / 128, AD_ / 80), 256, 0, stream>>>(X, W1, b1, h);

    // 3) X2 = h @ W2 + b2        (M=2048, K=400, N=400); block tile 128x80
    gemm_wmma_f32_kernel<AD_, AD_, 5, false>
        <<<dim3(B_ / 128, AD_ / 80), 256, 0, stream>>>(h, W2, b2, X2);

    // 4) fused gather-scores + softmax + entropy
    scores_softmax_kernel<<<B_, 256, 0, stream>>>(
        r_space, e_space, action_mask, X2, ent_emb, rel_emb, out_dist, out_entropy);
}